// SimpleGAT_31078383354330
// MI455X (gfx1250) — compile-verified
//
#include <hip/hip_runtime.h>

#define FD 128
#define LN_EPS 1e-5f

typedef float v2f __attribute__((ext_vector_type(2)));
typedef float v8f __attribute__((ext_vector_type(8)));

// Sum of all 128 values held as 4 floats/lane across the wave32.
// Uses V_WMMA_F32_16X16X4_F32 with B = ones: sum over a full column of D
// equals the sum of ALL A elements, independent of the lane<->(M,K) mapping,
// so we never need to know the exact A layout. Two WMMAs cover 128 values.
__device__ __forceinline__ float wave_total_wmma(float a0, float a1, float a2, float a3) {
    v2f A0; A0[0] = a0; A0[1] = a1;
    v2f A1; A1[0] = a2; A1[1] = a3;
    v2f B;  B[0] = 1.0f; B[1] = 1.0f;
    v8f C = {};
    C = __builtin_amdgcn_wmma_f32_16x16x4_f32(false, A0, false, B, (short)0, C, false, false);
    C = __builtin_amdgcn_wmma_f32_16x16x4_f32(false, A1, false, B, (short)0, C, false, false);
    float s = C[0] + C[1] + C[2] + C[3] + C[4] + C[5] + C[6] + C[7];
    // lanes 0-15 hold sum over M=0..7 of their column, lanes 16-31 hold M=8..15
    s += __shfl_xor(s, 16, 32);
    return s; // every lane: total of all 128 inputs
}

__global__ void __launch_bounds__(256) k_zero(int* __restrict__ cnt, int n) {
    int i = blockIdx.x * blockDim.x + threadIdx.x;
    if (i < n) cnt[i] = 0;
}

__global__ void __launch_bounds__(256) k_count(const int* __restrict__ dst,
                                               int* __restrict__ cnt, int E) {
    int i = blockIdx.x * blockDim.x + threadIdx.x;
    if (i < E) atomicAdd(&cnt[dst[i]], 1);
}

__global__ void __launch_bounds__(256) k_partial(const int* __restrict__ cnt,
                                                 int* __restrict__ part, int N) {
    __shared__ int sm[256];
    int i = blockIdx.x * 256 + threadIdx.x;
    sm[threadIdx.x] = (i < N) ? cnt[i] : 0;
    __syncthreads();
    for (int off = 128; off > 0; off >>= 1) {
        if (threadIdx.x < off) sm[threadIdx.x] += sm[threadIdx.x + off];
        __syncthreads();
    }
    if (threadIdx.x == 0) part[blockIdx.x] = sm[0];
}

// Exclusive scan of up to 1024 block partials in a single block.
__global__ void __launch_bounds__(1024) k_top(int* __restrict__ part, int nb,
                                              int* __restrict__ row_start, int N) {
    __shared__ int sm[1024];
    int t = threadIdx.x;
    int v = (t < nb) ? part[t] : 0;
    sm[t] = v;
    __syncthreads();
    for (int off = 1; off < 1024; off <<= 1) {
        int add = (t >= off) ? sm[t - off] : 0;
        __syncthreads();
        sm[t] += add;
        __syncthreads();
    }
    if (t < nb) part[t] = sm[t] - v;        // exclusive
    if (t == 0) row_start[N] = sm[1023];    // total edge count
}

__global__ void __launch_bounds__(256) k_apply(const int* __restrict__ cnt,
                                               const int* __restrict__ part,
                                               int* __restrict__ row_start,
                                               int* __restrict__ cursor, int N) {
    __shared__ int sm[256];
    int t = threadIdx.x;
    int i = blockIdx.x * 256 + t;
    int v = (i < N) ? cnt[i] : 0;
    sm[t] = v;
    __syncthreads();
    for (int off = 1; off < 256; off <<= 1) {
        int add = (t >= off) ? sm[t - off] : 0;
        __syncthreads();
        sm[t] += add;
        __syncthreads();
    }
    int excl = sm[t] - v + part[blockIdx.x];
    if (i < N) { row_start[i] = excl; cursor[i] = excl; }
}

__global__ void __launch_bounds__(256) k_scatter(const int* __restrict__ src,
                                                 const int* __restrict__ dst,
                                                 const float* __restrict__ val,
                                                 int* __restrict__ cursor,
                                                 int* __restrict__ ssrc,
                                                 float* __restrict__ sval, int E) {
    int i = blockIdx.x * blockDim.x + threadIdx.x;
    if (i >= E) return;
    int p = atomicAdd(&cursor[dst[i]], 1);
    ssrc[p] = src[i];
    sval[p] = val[i];
}

// One wave32 per destination node: register-resident aggregation + fused
// diagonal weight + LayerNorm (WMMA reductions) + optional ReLU.
__global__ void __launch_bounds__(256) k_agg(const float* __restrict__ src_feat,
                                             float* __restrict__ dst_feat,
                                             const int* __restrict__ row_start,
                                             const int* __restrict__ ssrc,
                                             const float* __restrict__ sval,
                                             const float* __restrict__ w,
                                             const float* __restrict__ gam,
                                             const float* __restrict__ bet,
                                             int n_nodes, int apply_relu) {
    const int lane = threadIdx.x & 31;
    const int node = (blockIdx.x * blockDim.x + threadIdx.x) >> 5;
    if (node >= n_nodes) return;                 // wave-uniform: EXEC stays full

    const int begin = row_start[node];
    const int end   = row_start[node + 1];

    float ax = 0.f, ay = 0.f, az = 0.f, aw = 0.f;
    const float* bp = src_feat + (size_t)lane * 4;

    for (int eb = begin; eb < end; eb += 32) {
        int idx = eb + lane;
        int s = 0; float v = 0.f;
        if (idx < end) { s = ssrc[idx]; v = sval[idx]; }
        int cnt = end - eb; if (cnt > 32) cnt = 32;
        #pragma unroll 4
        for (int i = 0; i < cnt; ++i) {
            int   si = __shfl(s, i, 32);
            float vi = __shfl(v, i, 32);
            const float4 xv = *(const float4*)(bp + (size_t)si * FD); // 512B coalesced
            ax = fmaf(vi, xv.x, ax);
            ay = fmaf(vi, xv.y, ay);
            az = fmaf(vi, xv.z, az);
            aw = fmaf(vi, xv.w, aw);
        }
    }

    // diagonal weight commutes with the edge sum: apply once per node
    const float4 wv = *(const float4*)(w + lane * 4);
    ax *= wv.x; ay *= wv.y; az *= wv.z; aw *= wv.w;

    // LayerNorm statistics via two independent WMMA reductions
    float sum = wave_total_wmma(ax, ay, az, aw);
    float sq  = wave_total_wmma(ax * ax, ay * ay, az * az, aw * aw);
    float mu  = sum * (1.0f / 128.0f);
    float var = sq * (1.0f / 128.0f) - mu * mu;
    var = fmaxf(var, 0.0f);
    float rstd = rsqrtf(var + LN_EPS);

    const float4 gv = *(const float4*)(gam + lane * 4);
    const float4 bv = *(const float4*)(bet + lane * 4);
    float4 o;
    o.x = (ax - mu) * rstd * gv.x + bv.x;
    o.y = (ay - mu) * rstd * gv.y + bv.y;
    o.z = (az - mu) * rstd * gv.z + bv.z;
    o.w = (aw - mu) * rstd * gv.w + bv.w;
    if (apply_relu) {
        o.x = fmaxf(o.x, 0.f); o.y = fmaxf(o.y, 0.f);
        o.z = fmaxf(o.z, 0.f); o.w = fmaxf(o.w, 0.f);
    }
    *(float4*)(dst_feat + (size_t)node * FD + (size_t)lane * 4) = o;
}

extern "C" void kernel_launch(void* const* d_in, const int* in_sizes, int n_in,
                              void* d_out, int out_size, void* d_ws, size_t ws_size,
                              hipStream_t stream) {
    const float* x    = (const float*)d_in[0];
    const int*   esrc = (const int*)d_in[1];
    const int*   edst = (const int*)d_in[2];
    const float* evv  = (const float*)d_in[3];
    const float* w1   = (const float*)d_in[4];
    const float* w2   = (const float*)d_in[5];
    const float* g1   = (const float*)d_in[6];
    const float* b1   = (const float*)d_in[7];
    const float* g2   = (const float*)d_in[8];
    const float* b2   = (const float*)d_in[9];
    float* out = (float*)d_out;

    const int N = in_sizes[0] / FD;
    const int E = in_sizes[1];

    // carve workspace (256B aligned)
    char* p = (char*)d_ws;
    auto carve = [&](size_t bytes) -> void* {
        void* r = (void*)p;
        p += (bytes + 255) & ~(size_t)255;
        return r;
    };
    float* h2        = (float*)carve((size_t)N * FD * sizeof(float));
    int*   cnt       = (int*)  carve((size_t)N * sizeof(int));
    int*   row_start = (int*)  carve((size_t)(N + 1) * sizeof(int));
    int*   cursor    = (int*)  carve((size_t)N * sizeof(int));
    int*   ssrc      = (int*)  carve((size_t)E * sizeof(int));
    float* sval      = (float*)carve((size_t)E * sizeof(float));
    const int NB = (N + 255) / 256;
    int*   part      = (int*)  carve((size_t)NB * sizeof(int));

    // build CSR-by-destination once (reused by both layers)
    k_zero   <<<(N + 255) / 256, 256, 0, stream>>>(cnt, N);
    k_count  <<<(E + 255) / 256, 256, 0, stream>>>(edst, cnt, E);
    k_partial<<<NB, 256, 0, stream>>>(cnt, part, N);
    k_top    <<<1, 1024, 0, stream>>>(part, NB, row_start, N);
    k_apply  <<<NB, 256, 0, stream>>>(cnt, part, row_start, cursor, N);
    k_scatter<<<(E + 255) / 256, 256, 0, stream>>>(esrc, edst, evv, cursor, ssrc, sval, E);

    // fused gather/aggregate/LN layers (one wave per node)
    const int agg_blocks = ((N * 32) + 255) / 256;
    k_agg<<<agg_blocks, 256, 0, stream>>>(x,  h2,  row_start, ssrc, sval, w1, g1, b1, N, 1);
    k_agg<<<agg_blocks, 256, 0, stream>>>(h2, out, row_start, ssrc, sval, w2, g2, b2, N, 0);
}